// CustomLayerMKM_14680198218164
// MI455X (gfx1250) — compile-verified
//
#include <hip/hip_runtime.h>
#include <stdint.h>

// CDNA5 / gfx1250 kernel: y = x @ kron(W2, W1)^T + bias via the factored form
//   Y_n(32x32) = W2(32x32) * X_n(32x32) * W1^T(32x32)   per token n
// All matrix math uses V_WMMA_F32_16X16X4_F32 (fp32, faithful to reference).
// Token staging uses the Tensor Data Mover (tensor_load_to_lds + s_wait_tensorcnt).

typedef __attribute__((ext_vector_type(2))) float    v2f;
typedef __attribute__((ext_vector_type(8))) float    v8f;
typedef __attribute__((ext_vector_type(4))) unsigned v4u;
typedef __attribute__((ext_vector_type(8))) int      v8i;
typedef __attribute__((ext_vector_type(4))) int      v4i;

#define WAVES      8
#define TPW        2                    // tokens per wave
#define TOK_PER_WG (WAVES * TPW)        // 16 tokens per workgroup
#define TOKSZ      1024                 // floats per token (32x32)

__global__ __launch_bounds__(256) void kron_wmma_kernel(
    const float* __restrict__ x,      // (N_TOK, 1024)
    const float* __restrict__ w1,     // (32, 32)  W1[o1][i1]
    const float* __restrict__ w2,     // (32, 32)  W2[f2][f1]
    const float* __restrict__ bias,   // (1024,)
    float* __restrict__ out)          // (N_TOK, 1024)
{
    __shared__ float lds_x[TOK_PER_WG * TOKSZ];   // 64 KB: X slots, reused for T

    const int lane = threadIdx.x & 31;
    const int wave = threadIdx.x >> 5;
    const int r    = lane & 15;              // M/N index within a 16-wide tile
    const int koff = (lane >> 4) << 1;       // 0 for lanes 0-15, 2 for lanes 16-31
    const size_t tok0 = (size_t)blockIdx.x * TOK_PER_WG + (size_t)wave * TPW;

    float* myx = lds_x + (size_t)(wave * TPW) * TOKSZ;

    // ---- constant weight fragments, resident in registers ----
    // Step 1 B operand: B[k=i1][n=o1] = W1^T, i.e. W1[n][k]
    // Step 2 A operand: A[m=f2][k=f1] = W2[m][k]
    v2f w1b[2][8];   // [n_tile][k_step]
    v2f w2a[2][8];   // [m_tile][k_step]
    #pragma unroll
    for (int t = 0; t < 2; ++t)
      #pragma unroll
      for (int ks = 0; ks < 8; ++ks) {
        const int k  = ks * 4 + koff;
        const int rn = t * 16 + r;
        w1b[t][ks] = *(const v2f*)(w1 + rn * 32 + k);   // W1[n][k], W1[n][k+1]
        w2a[t][ks] = *(const v2f*)(w2 + rn * 32 + k);   // W2[m][k], W2[m][k+1]
      }

    // Bias laid out exactly like a C-tile: element (M, N) -> bias[M*32 + N]
    v8f bf[2][2];
    #pragma unroll
    for (int mt = 0; mt < 2; ++mt)
      #pragma unroll
      for (int nt = 0; nt < 2; ++nt)
        #pragma unroll
        for (int i = 0; i < 8; ++i)
          bf[mt][nt][i] = bias[(mt * 16 + i + koff * 4) * 32 + nt * 16 + r];

    // ---- stage this wave's TPW tokens into LDS via the Tensor Data Mover ----
#if defined(__gfx1250__) && __has_builtin(__builtin_amdgcn_tensor_load_to_lds)
    {
      const unsigned long long ga = (unsigned long long)(uintptr_t)(x + tok0 * TOKSZ);
      const unsigned lds_off = (unsigned)(uintptr_t)myx;   // low 32 bits = LDS byte addr
      // D# group 0: count=1 | lds_addr | global_addr | type=2
      v4u g0 = { 1u,
                 lds_off,
                 (unsigned)(ga & 0xffffffffull),
                 (unsigned)(ga >> 32) | (2u << 30) };
      // D# group 1: data_size=4B; tensor_dim0=1024; tensor_dim1=TPW;
      //             tile_dim0=1024; tile_dim1=TPW; tensor_dim0_stride=1024
      v8i g1 = { (int)(2u << 16),            // [17:16] data_size = 2 (4 bytes)
                 (int)(1024u << 16),         // [63:48] tensor_dim0 lo16
                 (int)((unsigned)TPW << 16), // [95:80] tensor_dim1 lo16
                 (int)(1024u << 16),         // [127:112] tile_dim0
                 (int)TPW,                   // [143:128] tile_dim1
                 1024,                       // [191:160] tensor_dim0_stride lo32
                 0, 0 };
      v4i gz4 = { 0, 0, 0, 0 };                          // groups 2/3 unused (2D)
      v8i gz8 = { 0, 0, 0, 0, 0, 0, 0, 0 };              // extra group (clang-23 form)
      __builtin_amdgcn_tensor_load_to_lds(g0, g1, gz4, gz4, gz8, 0);
      __builtin_amdgcn_s_wait_tensorcnt(0);
    }
#else
    #pragma unroll
    for (int e = 0; e < (TPW * TOKSZ) / 64; ++e) {   // float2 per lane, coalesced
      const int idx = (e * 32 + lane) * 2;
      *(v2f*)(myx + idx) = *(const v2f*)(x + tok0 * TOKSZ + idx);
    }
#endif

    // ---- per-token: two chained 32x32x32 matmuls on the fp32 WMMA pipe ----
    for (int ti = 0; ti < TPW; ++ti) {
      float* X = myx + ti * TOKSZ;          // X_n, later overwritten with T_n

      // Step 1: T = X * W1^T   (A = X fragments from LDS, B = w1b registers)
      v8f acc[2][2];
      #pragma unroll
      for (int mt = 0; mt < 2; ++mt)
        #pragma unroll
        for (int nt = 0; nt < 2; ++nt)
          acc[mt][nt] = (v8f){0.f,0.f,0.f,0.f,0.f,0.f,0.f,0.f};

      #pragma unroll
      for (int mt = 0; mt < 2; ++mt)
        #pragma unroll
        for (int ks = 0; ks < 8; ++ks) {
          // A 16x4 fp32 fragment: lane holds X[m][k], X[m][k+1] (ds_load_b64)
          v2f a = *(const v2f*)(X + (mt * 16 + r) * 32 + ks * 4 + koff);
          #pragma unroll
          for (int nt = 0; nt < 2; ++nt)
            acc[mt][nt] = __builtin_amdgcn_wmma_f32_16x16x4_f32(
                false, a, false, w1b[nt][ks], (short)0, acc[mt][nt], false, false);
        }

      // Spill T into the (consumed) X slot; same-wave DS ops are in-order.
      #pragma unroll
      for (int mt = 0; mt < 2; ++mt)
        #pragma unroll
        for (int nt = 0; nt < 2; ++nt)
          #pragma unroll
          for (int i = 0; i < 8; ++i)
            X[(mt * 16 + i + koff * 4) * 32 + nt * 16 + r] = acc[mt][nt][i];

      // Step 2: Y = W2 * T + bias (A = w2a registers, B = T from LDS, C = bias)
      v8f o[2][2];
      #pragma unroll
      for (int mt = 0; mt < 2; ++mt)
        #pragma unroll
        for (int nt = 0; nt < 2; ++nt)
          o[mt][nt] = bf[mt][nt];

      #pragma unroll
      for (int ks = 0; ks < 8; ++ks)
        #pragma unroll
        for (int nt = 0; nt < 2; ++nt) {
          const int k = ks * 4 + koff;
          v2f b;
          b.x = X[k * 32       + nt * 16 + r];   // T[k][n]
          b.y = X[(k + 1) * 32 + nt * 16 + r];   // T[k+1][n]
          #pragma unroll
          for (int mt = 0; mt < 2; ++mt)
            o[mt][nt] = __builtin_amdgcn_wmma_f32_16x16x4_f32(
                false, w2a[mt][ks], false, b, (short)0, o[mt][nt], false, false);
        }

      // Store: C-tile (M=f2 rows, N=o1 cols) matches out[n, f2*32+o1] directly.
      float* dst = out + (tok0 + ti) * 1024;
      #pragma unroll
      for (int mt = 0; mt < 2; ++mt)
        #pragma unroll
        for (int nt = 0; nt < 2; ++nt)
          #pragma unroll
          for (int i = 0; i < 8; ++i)
            dst[(mt * 16 + i + koff * 4) * 32 + nt * 16 + r] = o[mt][nt][i];
    }
}

extern "C" void kernel_launch(void* const* d_in, const int* in_sizes, int n_in,
                              void* d_out, int out_size, void* d_ws, size_t ws_size,
                              hipStream_t stream) {
    const float* x    = (const float*)d_in[0];
    const float* w1   = (const float*)d_in[1];
    const float* w2   = (const float*)d_in[2];
    const float* bias = (const float*)d_in[3];
    float* out        = (float*)d_out;

    const int n_tok = in_sizes[0] / TOKSZ;          // 8 * 8192 = 65536
    const int grid  = n_tok / TOK_PER_WG;           // 4096 workgroups
    kron_wmma_kernel<<<grid, 256, 0, stream>>>(x, w1, w2, bias, out);
}